// MultiAffineFusor_90220083020031
// MI455X (gfx1250) — compile-verified
//
#include <hip/hip_runtime.h>

typedef __attribute__((ext_vector_type(2))) float v2f;
typedef __attribute__((ext_vector_type(8))) float v8f;

#define NB 8      // batch
#define NL 512    // seq len
#define ND 256    // in_dim
#define D1 257    // in_dim + bias
#define DP 260    // padded feature dim (multiple of 4, zero-filled 257..259)
#define NO 64     // out_dim

// out[b,l,m,o] = sum_i sum_j x2a[b,m,i] * U[o,i,j] * x1a[b,l,j]
// One workgroup: fixed (b, 16-l block, 16-o chunk). 256 threads = 8 waves.
__global__ __launch_bounds__(256) void MultiAffineFusor_kernel(
    const float* __restrict__ x1,   // (B, L, 1, D)
    const float* __restrict__ x2,   // (B, 1, L, D)
    const float* __restrict__ U,    // (O, D+1, D+1)
    float* __restrict__ out)        // (B, L, L, O)
{
  __shared__ float X1s[16 * DP];        // x1a tile: [l][j], bias at j=256
  __shared__ float Ts[16 * 16 * DP];    // T tile:   [l][oo][i]

  const int lblk = blockIdx.x;          // 0..31 : l block of 16
  const int o0   = blockIdx.y * 16;     // 0..48 : o chunk of 16
  const int b    = blockIdx.z;

  const int tid  = threadIdx.x;
  const int wave = tid >> 5;
  const int lane = tid & 31;
  const int half = lane >> 4;           // K sub-pair selector for f32 WMMA frags
  const int n16  = lane & 15;           // M/N index within 16

  const v8f vzero = {0.f, 0.f, 0.f, 0.f, 0.f, 0.f, 0.f, 0.f};

  // ---------------- load x1a tile into LDS (zero-padded, bias column) -------
  const float* x1b = x1 + ((size_t)b * NL + (size_t)lblk * 16) * ND;
  for (int idx = tid; idx < 16 * DP; idx += 256) {
    int l = idx / DP, j = idx % DP;
    float v = 0.0f;
    if (j < ND)       v = x1b[l * ND + j];
    else if (j == ND) v = 1.0f;          // bias feature
    X1s[idx] = v;
  }
  __syncthreads();

  // ---------------- stage A: Ts[l][oo][i] = sum_j X1s[l][j] * U[o,i,j] ------
  // Jobs = (oo, i-tile-pair): 16 * 8 = 128. Two independent accumulator
  // chains per job sharing one A-frag; fully branchless inner loop.
  for (int jid = wave; jid < 128; jid += 8) {
    const int oo  = jid >> 3;           // 0..15
    const int itp = jid & 7;            // 0..7 : pair of i-tiles (i < 256)
    const float* Uo = U + (size_t)(o0 + oo) * D1 * D1;
    const float* r0 = Uo + (size_t)(itp * 32 + n16) * D1;  // row i = itp*32+n16
    const float* r1 = r0 + 16 * (size_t)D1;                // row +16
    v8f c0 = vzero, c1 = vzero;
#pragma unroll 4
    for (int kt = 0; kt < 64; ++kt) {
      const int k = kt * 4 + half * 2;  // k+1 <= 255: always in range
      v2f a = *(const v2f*)&X1s[n16 * DP + k];
      v2f b0, b1;
      b0.x = r0[k]; b0.y = r0[k + 1];
      b1.x = r1[k]; b1.y = r1[k + 1];
      c0 = __builtin_amdgcn_wmma_f32_16x16x4_f32(false, a, false, b0,
                                                 (short)0, c0, false, false);
      c1 = __builtin_amdgcn_wmma_f32_16x16x4_f32(false, a, false, b1,
                                                 (short)0, c1, false, false);
    }
    { // k-tail: k = 256..259. A-frag from zero-padded LDS ({1,0} / {0,0});
      // U bias column loaded unconditionally, masked by cndmask (no branch).
      v2f a = *(const v2f*)&X1s[n16 * DP + 256 + half * 2];
      const float u0 = r0[256], u1 = r1[256];
      v2f b0, b1;
      b0.x = half ? 0.f : u0; b0.y = 0.f;
      b1.x = half ? 0.f : u1; b1.y = 0.f;
      c0 = __builtin_amdgcn_wmma_f32_16x16x4_f32(false, a, false, b0,
                                                 (short)0, c0, false, false);
      c1 = __builtin_amdgcn_wmma_f32_16x16x4_f32(false, a, false, b1,
                                                 (short)0, c1, false, false);
    }
    const int ibase = itp * 32 + n16;
#pragma unroll
    for (int r = 0; r < 8; ++r) {
      const int l = r + 8 * half;       // C layout: M = r + 8*(lane/16)
      float* p = &Ts[(l * 16 + oo) * DP + ibase];
      p[0]  = c0[r];
      p[16] = c1[r];
    }
  }

  // Bias row i=256 of T (scalar dot, 1 (l,oo) pair per thread) + zero pad.
  {
    const int l  = tid >> 4;            // 0..15
    const int oo = tid & 15;            // 0..15
    const float* Ur = U + ((size_t)(o0 + oo) * D1 + 256) * D1;  // row i = 256
    const float* xr = &X1s[l * DP];
    float s = 0.f;
    for (int j = 0; j < D1; ++j) s += xr[j] * Ur[j];
    float* trow = &Ts[(l * 16 + oo) * DP];
    trow[256] = s;
    trow[257] = 0.f; trow[258] = 0.f; trow[259] = 0.f;
  }
  __syncthreads();

  // ---------------- stage B: out[b,l,m,o] = sum_i x2a[b,m,i] * Ts[l][o][i] --
  // Jobs = (l-group of 4) x (32 m-tiles); 4 independent chains per A-frag.
  const float* x2b = x2 + (size_t)b * NL * ND;
  for (int jid = wave; jid < 4 * 32; jid += 8) {
    const int lg = jid >> 5;            // 0..3 : group of 4 l-rows
    const int mt = jid & 31;            // 0..31: m tile
    const float* xr = x2b + (size_t)(mt * 16 + n16) * ND;  // row m
    v8f acc[4];
#pragma unroll
    for (int q = 0; q < 4; ++q) acc[q] = vzero;

#pragma unroll 4
    for (int kt = 0; kt < 64; ++kt) {
      const int k = kt * 4 + half * 2;  // k+1 <= 255: always in range
      v2f a = *(const v2f*)&xr[k];      // 8B aligned -> one global_load_b64
#pragma unroll
      for (int q = 0; q < 4; ++q) {
        const int l = lg * 4 + q;
        v2f bb = *(const v2f*)&Ts[(l * 16 + n16) * DP + k];
        acc[q] = __builtin_amdgcn_wmma_f32_16x16x4_f32(false, a, false, bb,
                                                       (short)0, acc[q],
                                                       false, false);
      }
    }
    { // k-tail: x2a bias (i=256): A-frag is the constant {1,0} / {0,0};
      // Ts rows 256..259 are (s, 0, 0, 0) -> plain LDS reads, no branches.
      v2f a;
      a.x = half ? 0.f : 1.f; a.y = 0.f;
      const int k = 256 + half * 2;
#pragma unroll
      for (int q = 0; q < 4; ++q) {
        const int l = lg * 4 + q;
        v2f bb = *(const v2f*)&Ts[(l * 16 + n16) * DP + k];
        acc[q] = __builtin_amdgcn_wmma_f32_16x16x4_f32(false, a, false, bb,
                                                       (short)0, acc[q],
                                                       false, false);
      }
    }
    // store: N=o contiguous -> 64B coalesced per half-wave
#pragma unroll
    for (int q = 0; q < 4; ++q) {
      const int l = lblk * 16 + lg * 4 + q;
#pragma unroll
      for (int r = 0; r < 8; ++r) {
        const int mrow = mt * 16 + r + 8 * half;
        out[(((size_t)b * NL + l) * NL + mrow) * NO + (o0 + n16)] = acc[q][r];
      }
    }
  }
}

extern "C" void kernel_launch(void* const* d_in, const int* in_sizes, int n_in,
                              void* d_out, int out_size, void* d_ws, size_t ws_size,
                              hipStream_t stream) {
  (void)in_sizes; (void)n_in; (void)d_ws; (void)ws_size; (void)out_size;
  const float* x1 = (const float*)d_in[0];
  const float* x2 = (const float*)d_in[1];
  const float* U  = (const float*)d_in[2];
  float* out = (float*)d_out;
  dim3 grid(NL / 16, NO / 16, NB);      // 32 x 4 x 8 = 1024 workgroups
  MultiAffineFusor_kernel<<<grid, dim3(256), 0, stream>>>(x1, x2, U, out);
}